// HierarchicalSoftmax_31722628448839
// MI455X (gfx1250) — compile-verified
//
#include <hip/hip_runtime.h>
#include <math.h>

#define BATCH    2048
#define NHID     1024
#define NTPC     225
#define NCLASSES 224

typedef float v2f __attribute__((ext_vector_type(2)));
typedef float v8f __attribute__((ext_vector_type(8)));

// ---------------------------------------------------------------- utility ---
__global__ void zero_counts_kernel(int* __restrict__ counts) {
    int i = blockIdx.x * blockDim.x + threadIdx.x;
    if (i < NCLASSES) counts[i] = 0;
}

__global__ void prep_kernel(const int* __restrict__ labels,
                            int* __restrict__ pos_top, int* __restrict__ pos_bot,
                            int* __restrict__ counts) {
    int b = blockIdx.x * blockDim.x + threadIdx.x;
    if (b < BATCH) {
        int l = labels[b];
        int c = l / NTPC;
        pos_top[b] = c;
        pos_bot[b] = l % NTPC;
        atomicAdd(&counts[c], 1);
    }
}

__global__ void scan_kernel(const int* __restrict__ counts,
                            int* __restrict__ offsets, int* __restrict__ cursor) {
    if (threadIdx.x == 0) {
        int acc = 0;
        for (int c = 0; c < NCLASSES; ++c) {
            offsets[c] = acc;
            cursor[c]  = acc;
            acc += counts[c];
        }
        offsets[NCLASSES] = acc;
    }
}

__global__ void scatter_kernel(const int* __restrict__ pos_top,
                               int* __restrict__ cursor, int* __restrict__ perm) {
    int b = blockIdx.x * blockDim.x + threadIdx.x;
    if (b < BATCH) {
        int slot = atomicAdd(&cursor[pos_top[b]], 1);
        perm[slot] = b;   // slot order is nondeterministic, values are not
    }
}

// ------------------------------------------------------------- top GEMM -----
// out[2048,224] = X[2048,1024] @ W[1024,224] + bias ; one wave per 16x16 tile.
__global__ void top_gemm_kernel(const float* __restrict__ X,
                                const float* __restrict__ W,
                                const float* __restrict__ bias,
                                float* __restrict__ out) {
    int lane = threadIdx.x & 31;
    int wave = threadIdx.x >> 5;
    int tile = blockIdx.x * 8 + wave;       // 224 blocks * 8 waves = 1792 tiles
    int mt = tile / 14;                     // 128 M-tiles
    int nt = tile % 14;                     // 14  N-tiles

    int r16   = lane & 15;
    int khalf = (lane >> 4) * 2;            // 0 or 2
    const float* Arow = X + (size_t)(mt * 16 + r16) * NHID;
    int col = nt * 16 + r16;

    v8f acc = {};
#pragma unroll 4
    for (int k0 = 0; k0 < NHID; k0 += 4) {
        int kA = k0 + khalf;
        v2f a = *(const v2f*)(Arow + kA);                 // K=kA, kA+1 (8B aligned)
        v2f b;
        b.x = W[(size_t)kA       * NCLASSES + col];
        b.y = W[(size_t)(kA + 1) * NCLASSES + col];
        acc = __builtin_amdgcn_wmma_f32_16x16x4_f32(false, a, false, b,
                                                    (short)0, acc, false, false);
    }
    // D layout: VGPR r -> M = r + 8*(lane>=16), N = lane&15
    int cn    = nt * 16 + r16;
    int mbase = mt * 16 + ((lane >> 4) << 3);
    float bv  = bias[cn];
#pragma unroll
    for (int r = 0; r < 8; ++r)
        out[(size_t)(mbase + r) * NCLASSES + cn] = acc[r] + bv;
}

// ---------------------------------------------------------- bottom GEMM -----
// For each class c: rows = samples of class c (via perm), GEMM vs bottom_W[c].
// grid = (15 N-tiles, 224 classes), one wave per block.
__global__ void bot_gemm_kernel(const float* __restrict__ X,
                                const float* __restrict__ BW,
                                const float* __restrict__ Bb,
                                const int* __restrict__ offsets,
                                const int* __restrict__ perm,
                                float* __restrict__ out) {
    int lane = threadIdx.x & 31;
    int c  = blockIdx.y;
    int nt = blockIdx.x;                     // 0..14 (N tiles, 225 cols)
    int r16   = lane & 15;
    int khalf = (lane >> 4) * 2;

    int colLane = nt * 16 + r16;
    int colC    = colLane < NTPC ? colLane : NTPC - 1;   // clamp (unused cols not stored)
    int start = offsets[c];
    int n     = offsets[c + 1] - start;
    const float* Wc = BW + (size_t)c * NHID * NTPC;

    for (int m0 = 0; m0 < n; m0 += 16) {
        int mrow = m0 + r16;
        int samp = (mrow < n) ? perm[start + mrow] : perm[start];  // pad rows reuse a valid sample
        const float* Arow = X + (size_t)samp * NHID;

        v8f acc = {};
#pragma unroll 4
        for (int k0 = 0; k0 < NHID; k0 += 4) {
            int kA = k0 + khalf;
            v2f a = *(const v2f*)(Arow + kA);
            v2f b;
            b.x = Wc[(size_t)kA       * NTPC + colC];
            b.y = Wc[(size_t)(kA + 1) * NTPC + colC];
            acc = __builtin_amdgcn_wmma_f32_16x16x4_f32(false, a, false, b,
                                                        (short)0, acc, false, false);
        }
        int  mOff = (lane >> 4) << 3;
        bool colOK = colLane < NTPC;
        float bv = Bb[(size_t)c * NTPC + colC];
#pragma unroll
        for (int r = 0; r < 8; ++r) {
            int mr = m0 + mOff + r;
            if (mr < n && colOK) {
                int s = perm[start + mr];
                out[(size_t)s * NTPC + colLane] = acc[r] + bv;
            }
        }
    }
}

// ----------------------------------------------- softmax + argmax (rows) ----
__global__ void softmax_argmax_kernel(float* __restrict__ mat, int ncols,
                                      float* __restrict__ argout) {
    int row = blockIdx.x;
    float* rowp = mat + (size_t)row * ncols;
    int t = threadIdx.x;
    __shared__ float smax[256];
    __shared__ int   sidx[256];
    __shared__ float ssum[256];

    float vmax = -INFINITY; int vidx = 0x7fffffff;
    for (int cidx = t; cidx < ncols; cidx += 256) {
        float v = rowp[cidx];
        if (v > vmax) { vmax = v; vidx = cidx; }
    }
    smax[t] = vmax; sidx[t] = vidx;
    __syncthreads();
    for (int s = 128; s > 0; s >>= 1) {
        if (t < s) {
            float o = smax[t + s]; int oi = sidx[t + s];
            if (o > smax[t] || (o == smax[t] && oi < sidx[t])) { smax[t] = o; sidx[t] = oi; }
        }
        __syncthreads();
    }
    float rmax = smax[0];
    float lsum = 0.f;
    for (int cidx = t; cidx < ncols; cidx += 256) {
        float e = expf(rowp[cidx] - rmax);
        rowp[cidx] = e;
        lsum += e;
    }
    ssum[t] = lsum;
    __syncthreads();
    for (int s = 128; s > 0; s >>= 1) {
        if (t < s) ssum[t] += ssum[t + s];
        __syncthreads();
    }
    float inv = 1.0f / ssum[0];
    for (int cidx = t; cidx < ncols; cidx += 256) rowp[cidx] *= inv;
    if (t == 0) argout[row] = (float)sidx[0];
}

// ------------------------------------------------------------- finalize -----
__global__ void finalize_kernel(const float* __restrict__ topp,
                                const float* __restrict__ botp,
                                const int* __restrict__ pos_top,
                                const int* __restrict__ pos_bot,
                                const float* __restrict__ tidx,
                                const float* __restrict__ bidx,
                                float* __restrict__ out_tp,
                                float* __restrict__ out_ridx,
                                float* __restrict__ logp) {
    int b = blockIdx.x * blockDim.x + threadIdx.x;
    if (b < BATCH) {
        float tp = topp[(size_t)b * NCLASSES + pos_top[b]]
                 * botp[(size_t)b * NTPC     + pos_bot[b]];
        out_tp[b]   = tp;
        logp[b]     = logf(tp);
        out_ridx[b] = tidx[b] * (float)NTPC + bidx[b];
    }
}

__global__ void loss_kernel(const float* __restrict__ logp, float* __restrict__ out_loss) {
    __shared__ float s[256];
    int t = threadIdx.x;
    float acc = 0.f;
    for (int b = t; b < BATCH; b += 256) acc += logp[b];  // fixed order -> deterministic
    s[t] = acc;
    __syncthreads();
    for (int st = 128; st > 0; st >>= 1) {
        if (t < st) s[t] += s[t + st];
        __syncthreads();
    }
    if (t == 0) out_loss[0] = -s[0] / (float)BATCH;
}

// --------------------------------------------------------------- launch -----
extern "C" void kernel_launch(void* const* d_in, const int* in_sizes, int n_in,
                              void* d_out, int out_size, void* d_ws, size_t ws_size,
                              hipStream_t stream) {
    const float* inputs   = (const float*)d_in[0];
    const int*   labels   = (const int*)  d_in[1];
    const float* top_W    = (const float*)d_in[2];
    const float* top_b    = (const float*)d_in[3];
    const float* bottom_W = (const float*)d_in[4];
    const float* bottom_b = (const float*)d_in[5];

    // workspace layout
    int* counts  = (int*)d_ws;              // 224
    int* cursor  = counts  + NCLASSES;      // 224
    int* offsets = cursor  + NCLASSES;      // 225
    int* pos_top = offsets + NCLASSES + 1;  // 2048
    int* pos_bot = pos_top + BATCH;         // 2048
    int* perm    = pos_bot + BATCH;         // 2048
    float* logp  = (float*)(perm + BATCH);  // 2048

    // output layout (tuple flattened in return order, all as float)
    float* out     = (float*)d_out;
    float* o_loss  = out;                               // 1
    float* o_tp    = o_loss + 1;                        // 2048
    float* o_topp  = o_tp   + BATCH;                    // 2048*224 (logits->probs in place)
    float* o_botp  = o_topp + (size_t)BATCH * NCLASSES; // 2048*225 (logits->probs in place)
    float* o_tidx  = o_botp + (size_t)BATCH * NTPC;     // 2048
    float* o_bidx  = o_tidx + BATCH;                    // 2048
    float* o_ridx  = o_bidx + BATCH;                    // 2048

    zero_counts_kernel<<<1, 256, 0, stream>>>(counts);
    prep_kernel<<<BATCH / 256, 256, 0, stream>>>(labels, pos_top, pos_bot, counts);
    scan_kernel<<<1, 32, 0, stream>>>(counts, offsets, cursor);
    scatter_kernel<<<BATCH / 256, 256, 0, stream>>>(pos_top, cursor, perm);

    // top GEMM: 128*14 = 1792 tiles, 8 waves/block -> 224 blocks
    top_gemm_kernel<<<224, 256, 0, stream>>>(inputs, top_W, top_b, o_topp);

    // bottom GEMM: 15 N-tiles x 224 classes, one wave each
    bot_gemm_kernel<<<dim3(15, NCLASSES), 32, 0, stream>>>(inputs, bottom_W, bottom_b,
                                                           offsets, perm, o_botp);

    softmax_argmax_kernel<<<BATCH, 256, 0, stream>>>(o_topp, NCLASSES, o_tidx);
    softmax_argmax_kernel<<<BATCH, 256, 0, stream>>>(o_botp, NTPC, o_bidx);

    finalize_kernel<<<BATCH / 256, 256, 0, stream>>>(o_topp, o_botp, pos_top, pos_bot,
                                                     o_tidx, o_bidx, o_tp, o_ridx, logp);
    loss_kernel<<<1, 256, 0, stream>>>(logp, o_loss);
}